// MolecularInteractionFusionModule_81999515615432
// MI455X (gfx1250) — compile-verified
//
#include <hip/hip_runtime.h>
#include <math.h>

typedef __attribute__((ext_vector_type(2))) float v2f;
typedef __attribute__((ext_vector_type(8))) float v8f;

#define CDIM 128

// ---------------- utility ----------------

__global__ void k_fill(float* __restrict__ p, float v, long long n) {
    long long i = (long long)blockIdx.x * blockDim.x + threadIdx.x;
    long long stride = (long long)gridDim.x * blockDim.x;
    for (; i < n; i += stride) p[i] = v;
}

__device__ __forceinline__ void atomic_max_f32(float* addr, float val) {
    if (val >= 0.f)
        atomicMax((int*)addr, __float_as_int(val));
    else
        atomicMin((unsigned int*)addr, __float_as_uint(val));
}

// ---------------- fp32 WMMA GEMM: OUT[M,N] = A[M,K] @ B[K,N] (+bias) (+act) ----------------
// REQUIRES M % 32 == 0 and N % 32 == 0 (true for every call in this net:
// M in {20000,256}, N in {512,640,128,384}). K remainder handled by guarded tail.
// One wave computes a 32x32 tile: 2 A-frags x 2 B-frags -> 4 WMMAs per K-step,
// with no EXEC-mask manipulation anywhere in the hot loop.
// act: 0 = none, 1 = relu
__global__ void k_gemm_wmma(const float* __restrict__ A, const float* __restrict__ B,
                            const float* __restrict__ bias, float* __restrict__ OUT,
                            int M, int N, int K, int act) {
    int gid  = blockIdx.x * blockDim.x + threadIdx.x;
    int wave = gid >> 5;
    int lane = threadIdx.x & 31;
    int totalWaves = (gridDim.x * blockDim.x) >> 5;
    int tilesN = N >> 5;
    int tilesM = M >> 5;
    int nTiles = tilesM * tilesN;
    int mloc = lane & 15;          // sub-row within 16x16 (A) / sub-col (B)
    int koff = (lane >> 4) << 1;   // 0 or 2
    int K4 = K & ~3;

    for (int t = wave; t < nTiles; t += totalWaves) {
        int tm = t / tilesN, tn = t - tm * tilesN;
        int row0 = tm << 5, col0 = tn << 5;
        long long arow0 = (long long)(row0 + mloc) * K;
        long long arow1 = arow0 + 16LL * K;
        int n0 = col0 + mloc;
        int n1 = n0 + 16;

        v8f acc00 = {}, acc01 = {}, acc10 = {}, acc11 = {};

        #pragma unroll 2
        for (int kk = 0; kk < K4; kk += 4) {
            int k0 = kk + koff;
            long long brow0 = (long long)k0 * N;
            long long brow1 = brow0 + N;
            v2f a0 = *(const v2f*)(A + arow0 + k0);
            v2f a1 = *(const v2f*)(A + arow1 + k0);
            v2f b0, b1v;
            b0.x  = B[brow0 + n0];
            b0.y  = B[brow1 + n0];
            b1v.x = B[brow0 + n1];
            b1v.y = B[brow1 + n1];
            acc00 = __builtin_amdgcn_wmma_f32_16x16x4_f32(false, a0, false, b0,
                                                          (short)0, acc00, false, false);
            acc01 = __builtin_amdgcn_wmma_f32_16x16x4_f32(false, a0, false, b1v,
                                                          (short)0, acc01, false, false);
            acc10 = __builtin_amdgcn_wmma_f32_16x16x4_f32(false, a1, false, b0,
                                                          (short)0, acc10, false, false);
            acc11 = __builtin_amdgcn_wmma_f32_16x16x4_f32(false, a1, false, b1v,
                                                          (short)0, acc11, false, false);
        }
        if (K4 < K) {  // guarded tail (only the K=954 cell GEMM reaches this)
            int k0 = K4 + koff;
            bool v0 = k0 < K, v1 = (k0 + 1) < K;
            long long brow0 = (long long)k0 * N;
            long long brow1 = brow0 + N;
            v2f a0, a1, b0, b1v;
            a0.x  = v0 ? A[arow0 + k0] : 0.f;
            a0.y  = v1 ? A[arow0 + k0 + 1] : 0.f;
            a1.x  = v0 ? A[arow1 + k0] : 0.f;
            a1.y  = v1 ? A[arow1 + k0 + 1] : 0.f;
            b0.x  = v0 ? B[brow0 + n0] : 0.f;
            b0.y  = v1 ? B[brow1 + n0] : 0.f;
            b1v.x = v0 ? B[brow0 + n1] : 0.f;
            b1v.y = v1 ? B[brow1 + n1] : 0.f;
            acc00 = __builtin_amdgcn_wmma_f32_16x16x4_f32(false, a0, false, b0,
                                                          (short)0, acc00, false, false);
            acc01 = __builtin_amdgcn_wmma_f32_16x16x4_f32(false, a0, false, b1v,
                                                          (short)0, acc01, false, false);
            acc10 = __builtin_amdgcn_wmma_f32_16x16x4_f32(false, a1, false, b0,
                                                          (short)0, acc10, false, false);
            acc11 = __builtin_amdgcn_wmma_f32_16x16x4_f32(false, a1, false, b1v,
                                                          (short)0, acc11, false, false);
        }

        // epilogue: C/D layout — VGPR r: lanes 0-15 -> M=r, lanes 16-31 -> M=r+8, N=lane&15
        int rsub = (lane >> 4) << 3;
        int c0 = col0 + mloc;
        float bc0 = bias ? bias[c0] : 0.f;
        float bc1 = bias ? bias[c0 + 16] : 0.f;
        #pragma unroll
        for (int r = 0; r < 8; ++r) {
            long long rr0 = (long long)(row0 + rsub + r) * N;
            long long rr1 = rr0 + 16LL * N;
            float v00 = acc00[r] + bc0;
            float v01 = acc01[r] + bc1;
            float v10 = acc10[r] + bc0;
            float v11 = acc11[r] + bc1;
            if (act == 1) {
                v00 = v00 > 0.f ? v00 : 0.f;
                v01 = v01 > 0.f ? v01 : 0.f;
                v10 = v10 > 0.f ? v10 : 0.f;
                v11 = v11 > 0.f ? v11 : 0.f;
            }
            OUT[rr0 + c0]      = v00;
            OUT[rr0 + c0 + 16] = v01;
            OUT[rr1 + c0]      = v10;
            OUT[rr1 + c0 + 16] = v11;
        }
    }
}

// ---------------- drug linear: x = float(triplets[:, :4]) @ Wd + bd ----------------
__global__ void k_drug_linear(const int* __restrict__ trip, const float* __restrict__ Wd,
                              const float* __restrict__ bd, float* __restrict__ x, int N) {
    int i = blockIdx.x * blockDim.x + threadIdx.x;
    if (i >= N * CDIM) return;
    int n = i >> 7, c = i & (CDIM - 1);
    float s = bd[c];
    for (int j = 0; j < 4; ++j) s += (float)trip[n * 4 + j] * Wd[j * CDIM + c];
    x[i] = s;
}

// ---------------- attention coefficients: es[n,h] = <xh[n,h,:], a_s[h,:]> ----------------
__global__ void k_attn_coef(const float* __restrict__ xh, const float* __restrict__ as_,
                            const float* __restrict__ ad_, float* __restrict__ es,
                            float* __restrict__ ed, int N, int H) {
    int gid = blockIdx.x * blockDim.x + threadIdx.x;
    int wave = gid >> 5, lane = threadIdx.x & 31;
    int total = N * H;
    int waves = (gridDim.x * blockDim.x) >> 5;
    for (int idx = wave; idx < total; idx += waves) {
        int n = idx / H, h = idx - n * H;
        const float* xr = xh + (long long)n * H * CDIM + h * CDIM;
        float s1 = 0.f, s2 = 0.f;
        for (int c = lane; c < CDIM; c += 32) {
            float v = xr[c];
            s1 += v * as_[h * CDIM + c];
            s2 += v * ad_[h * CDIM + c];
        }
        for (int off = 16; off > 0; off >>= 1) {
            s1 += __shfl_down(s1, off);
            s2 += __shfl_down(s2, off);
        }
        if (lane == 0) { es[idx] = s1; ed[idx] = s2; }
    }
}

// ---------------- edge pass 1: segment max ----------------
__global__ void k_edge_max(const int* __restrict__ src, const int* __restrict__ dst,
                           const float* __restrict__ es, const float* __restrict__ ed,
                           float* __restrict__ m, int E, int N, int H) {
    int i = blockIdx.x * blockDim.x + threadIdx.x;
    int total = E + N;
    if (i >= total) return;
    int s, d;
    if (i < E) { s = src[i]; d = dst[i]; if (s == d) return; }  // masked; self loops appended
    else { s = d = i - E; }
    for (int h = 0; h < H; ++h) {
        float e = es[s * H + h] + ed[d * H + h];
        e = e > 0.f ? e : 0.2f * e;
        atomic_max_f32(&m[d * H + h], e);
    }
}

// ---------------- edge pass 2: p = exp(e - m[dst]); segment sum ----------------
__global__ void k_edge_exp(const int* __restrict__ src, const int* __restrict__ dst,
                           const float* __restrict__ es, const float* __restrict__ ed,
                           const float* __restrict__ m, float* __restrict__ p,
                           float* __restrict__ ssum, int E, int N, int H) {
    int i = blockIdx.x * blockDim.x + threadIdx.x;
    int total = E + N;
    if (i >= total) return;
    int s, d;
    if (i < E) {
        s = src[i]; d = dst[i];
        if (s == d) { for (int h = 0; h < H; ++h) p[(long long)i * H + h] = 0.f; return; }
    } else { s = d = i - E; }
    for (int h = 0; h < H; ++h) {
        float e = es[s * H + h] + ed[d * H + h];
        e = e > 0.f ? e : 0.2f * e;
        float pv = __expf(e - m[d * H + h]);
        p[(long long)i * H + h] = pv;
        atomicAdd(&ssum[d * H + h], pv);
    }
}

// ---------------- edge pass 3: out[dst] += alpha * xh[src] (one wave per edge) ----------------
__global__ void k_edge_aggr(const int* __restrict__ src, const int* __restrict__ dst,
                            const float* __restrict__ p, const float* __restrict__ ssum,
                            const float* __restrict__ xh, float* __restrict__ out,
                            int E, int N, int H) {
    int gid = blockIdx.x * blockDim.x + threadIdx.x;
    int wave = gid >> 5, lane = threadIdx.x & 31;
    int total = E + N;
    if (wave >= total) return;
    int i = wave, s, d;
    if (i < E) { s = src[i]; d = dst[i]; if (s == d) return; }
    else { s = d = i - E; }
    float al[8];
    for (int h = 0; h < H; ++h) al[h] = p[(long long)i * H + h] / ssum[d * H + h];
    int HC = H * CDIM;
    const float* xr = xh + (long long)s * HC;
    float* orow = out + (long long)d * HC;
    for (int j = lane; j < HC; j += 32)
        atomicAdd(&orow[j], xr[j] * al[j >> 7]);
}

// ---------------- elu(acc + bias), in-place ----------------
__global__ void k_elu_bias(float* __restrict__ buf, const float* __restrict__ bias,
                           long long n, int HC) {
    long long i = (long long)blockIdx.x * blockDim.x + threadIdx.x;
    long long stride = (long long)gridDim.x * blockDim.x;
    for (; i < n; i += stride) {
        int j = (int)(i % HC);
        float v = buf[i] + bias[j];
        buf[i] = v > 0.f ? v : (__expf(v) - 1.f);
    }
}

// ---------------- readout: hmean[c] = mean_n (h1 + h2) ----------------
__global__ void k_readout_mean(const float* __restrict__ h1, const float* __restrict__ h2,
                               float* __restrict__ hmean, int N) {
    __shared__ float red[256];
    int c = blockIdx.x;
    float s = 0.f;
    for (int n = threadIdx.x; n < N; n += blockDim.x)
        s += h1[(long long)n * CDIM + c] + h2[(long long)n * CDIM + c];
    red[threadIdx.x] = s;
    __syncthreads();
    for (int off = 128; off > 0; off >>= 1) {
        if ((int)threadIdx.x < off) red[threadIdx.x] += red[threadIdx.x + off];
        __syncthreads();
    }
    if (threadIdx.x == 0) hmean[c] = red[0] / (float)N;
}

// ---------------- head MLP: hg = relu(h@Wfc+bfc)@Whg+bhg ; v_ = relu(hg@Wv+bv) ----------------
__global__ void k_head_mlp(const float* __restrict__ hmean, const float* __restrict__ Wfc,
                           const float* __restrict__ bfc, const float* __restrict__ Whg,
                           const float* __restrict__ bhg, const float* __restrict__ Wv,
                           const float* __restrict__ bv, float* __restrict__ vbar) {
    __shared__ float h[128], t1[256], hg[128];
    int t = threadIdx.x;  // 384 threads
    if (t < 128) h[t] = hmean[t];
    __syncthreads();
    if (t < 256) {
        float s = bfc[t];
        for (int c = 0; c < 128; ++c) s += h[c] * Wfc[c * 256 + t];
        t1[t] = s > 0.f ? s : 0.f;
    }
    __syncthreads();
    if (t < 128) {
        float s = bhg[t];
        for (int j = 0; j < 256; ++j) s += t1[j] * Whg[j * 128 + t];
        hg[t] = s;
    }
    __syncthreads();
    float s = bv[t];
    for (int c = 0; c < 128; ++c) s += hg[c] * Wv[c * 384 + t];
    vbar[t] = s > 0.f ? s : 0.f;
}

// ---------------- BAN: logits[b,c] = (sum_h att[b,h]) * sum_{u<3} v_[3c+u] q_[b,3c+u] ----------------
__global__ void k_ban_logits(const float* __restrict__ vbar, const float* __restrict__ q_,
                             const float* __restrict__ h_mat, const float* __restrict__ h_bias,
                             float* __restrict__ logits, int HOUT) {
    __shared__ float red[128];
    int b = blockIdx.x, t = threadIdx.x;  // 128 threads
    const float* qb = q_ + (long long)b * 384;
    float part = 0.f;
    for (int k = t; k < 384; k += 128) {
        float vq = vbar[k] * qb[k];
        float cs = 0.f;
        for (int h = 0; h < HOUT; ++h) cs += h_mat[h * 384 + k];
        part += cs * vq;
    }
    red[t] = part;
    __syncthreads();
    for (int off = 64; off > 0; off >>= 1) {
        if (t < off) red[t] += red[t + off];
        __syncthreads();
    }
    float hb = 0.f;
    for (int h = 0; h < HOUT; ++h) hb += h_bias[h];
    float attsum = red[0] + hb;
    float s = 0.f;
    for (int u = 0; u < 3; ++u) { int k = t * 3 + u; s += vbar[k] * qb[k]; }
    logits[(long long)b * CDIM + t] = attsum * s;
}

// ---------------- batch norm over batch dimension ----------------
__global__ void k_batchnorm(const float* __restrict__ logits, const float* __restrict__ gamma,
                            const float* __restrict__ beta, float* __restrict__ out, int B) {
    __shared__ float rs[256], rs2[256];
    int c = blockIdx.x;
    float s = 0.f, s2 = 0.f;
    for (int b = threadIdx.x; b < B; b += blockDim.x) {
        float v = logits[(long long)b * CDIM + c];
        s += v; s2 += v * v;
    }
    rs[threadIdx.x] = s; rs2[threadIdx.x] = s2;
    __syncthreads();
    for (int off = 128; off > 0; off >>= 1) {
        if ((int)threadIdx.x < off) {
            rs[threadIdx.x] += rs[threadIdx.x + off];
            rs2[threadIdx.x] += rs2[threadIdx.x + off];
        }
        __syncthreads();
    }
    float mu = rs[0] / (float)B;
    float var = rs2[0] / (float)B - mu * mu;
    float inv = rsqrtf(var + 1e-5f) * gamma[c];
    for (int b = threadIdx.x; b < B; b += blockDim.x) {
        float v = logits[(long long)b * CDIM + c];
        out[(long long)b * CDIM + c] = (v - mu) * inv + beta[c];
    }
}

// ---------------- GAT layer driver ----------------
static void run_gat(hipStream_t stream, const float* xin, const float* W, const float* as_,
                    const float* ad_, const float* bgat, const float* Wl, const float* bl,
                    float* xh, float* gacc, float* es, float* ed, float* mbuf, float* ssum,
                    float* pbuf, float* hout, const int* src, const int* dst,
                    int N, int E, int H) {
    int HC = H * CDIM;
    dim3 gB(2048), tB(256);
    k_gemm_wmma<<<gB, tB, 0, stream>>>(xin, W, nullptr, xh, N, HC, CDIM, 0);
    int nwaves = N * H;
    k_attn_coef<<<(nwaves * 32 + 255) / 256, 256, 0, stream>>>(xh, as_, ad_, es, ed, N, H);
    k_fill<<<256, 256, 0, stream>>>(mbuf, -1e30f, (long long)N * H);
    k_fill<<<256, 256, 0, stream>>>(ssum, 0.f, (long long)N * H);
    int tot = E + N;
    k_edge_max<<<(tot + 255) / 256, 256, 0, stream>>>(src, dst, es, ed, mbuf, E, N, H);
    k_edge_exp<<<(tot + 255) / 256, 256, 0, stream>>>(src, dst, es, ed, mbuf, pbuf, ssum, E, N, H);
    k_fill<<<2048, 256, 0, stream>>>(gacc, 0.f, (long long)N * HC);
    long long aggrThreads = (long long)tot * 32;
    k_edge_aggr<<<(int)((aggrThreads + 255) / 256), 256, 0, stream>>>(src, dst, pbuf, ssum, xh,
                                                                     gacc, E, N, H);
    k_elu_bias<<<2048, 256, 0, stream>>>(gacc, bgat, (long long)N * HC, HC);
    k_gemm_wmma<<<gB, tB, 0, stream>>>(gacc, Wl, bl, hout, N, CDIM, HC, 0);
}

// ---------------- entry point ----------------
extern "C" void kernel_launch(void* const* d_in, const int* in_sizes, int n_in,
                              void* d_out, int out_size, void* d_ws, size_t ws_size,
                              hipStream_t stream) {
    const float* cell = (const float*)d_in[0];
    const float* Wd   = (const float*)d_in[1];
    const float* bd   = (const float*)d_in[2];
    const float* W1   = (const float*)d_in[3];
    const float* a1s  = (const float*)d_in[4];
    const float* a1d  = (const float*)d_in[5];
    const float* b1   = (const float*)d_in[6];
    const float* Wl1  = (const float*)d_in[7];
    const float* bl1  = (const float*)d_in[8];
    const float* W2   = (const float*)d_in[9];
    const float* a2s  = (const float*)d_in[10];
    const float* a2d  = (const float*)d_in[11];
    const float* b2   = (const float*)d_in[12];
    const float* Wl2  = (const float*)d_in[13];
    const float* bl2  = (const float*)d_in[14];
    const float* Wfc  = (const float*)d_in[15];
    const float* bfc  = (const float*)d_in[16];
    const float* Whg  = (const float*)d_in[17];
    const float* bhg  = (const float*)d_in[18];
    const float* Wc   = (const float*)d_in[19];
    const float* bc   = (const float*)d_in[20];
    const float* Wv   = (const float*)d_in[21];
    const float* bv   = (const float*)d_in[22];
    const float* Wq   = (const float*)d_in[23];
    const float* bq   = (const float*)d_in[24];
    const float* hmat = (const float*)d_in[25];
    const float* hbias= (const float*)d_in[26];
    const float* gam  = (const float*)d_in[27];
    const float* bet  = (const float*)d_in[28];
    const int* trip   = (const int*)d_in[29];
    const int* eidx   = (const int*)d_in[30];

    int N = in_sizes[29] / 4;
    int E = in_sizes[30] / 2;
    int B = in_sizes[0] / 954;
    int HOUT = in_sizes[26];
    int CF = 954;
    const int* src = eidx;
    const int* dst = eidx + E;

    float* w = (float*)d_ws;
    size_t o = 0;
    float* x     = w + o; o += (size_t)N * CDIM;
    float* xh    = w + o; o += (size_t)N * 640;      // max of 4*128 / 5*128
    float* gacc  = w + o; o += (size_t)N * 640;
    float* h1    = w + o; o += (size_t)N * CDIM;
    float* h2    = w + o; o += (size_t)N * CDIM;
    float* es    = w + o; o += (size_t)N * 5;
    float* edc   = w + o; o += (size_t)N * 5;
    float* mbuf  = w + o; o += (size_t)N * 5;
    float* ssum  = w + o; o += (size_t)N * 5;
    float* pbuf  = w + o; o += (size_t)(E + N) * 5;
    float* hmean = w + o; o += 128;
    float* vbar  = w + o; o += 384;
    float* cbuf  = w + o; o += (size_t)B * CDIM;
    float* qbuf  = w + o; o += (size_t)B * 384;
    float* logit = w + o; o += (size_t)B * CDIM;
    (void)ws_size; (void)n_in; (void)out_size;

    // 1) drug linear
    k_drug_linear<<<(N * CDIM + 255) / 256, 256, 0, stream>>>(trip, Wd, bd, x, N);

    // 2) GAT layer 1 (H=4): x -> h1
    run_gat(stream, x, W1, a1s, a1d, b1, Wl1, bl1, xh, gacc, es, edc, mbuf, ssum, pbuf,
            h1, src, dst, N, E, 4);

    // 3) GAT layer 2 (H=5): h1 -> h2
    run_gat(stream, h1, W2, a2s, a2d, b2, Wl2, bl2, xh, gacc, es, edc, mbuf, ssum, pbuf,
            h2, src, dst, N, E, 5);

    // 4) graph readout
    k_readout_mean<<<CDIM, 256, 0, stream>>>(h1, h2, hmean, N);

    // 5) head MLP -> v_
    k_head_mlp<<<1, 384, 0, stream>>>(hmean, Wfc, bfc, Whg, bhg, Wv, bv, vbar);

    // 6) cell branch: c = relu(cell @ Wc + bc)   [B,954]x[954,128]
    k_gemm_wmma<<<256, 256, 0, stream>>>(cell, Wc, bc, cbuf, B, CDIM, CF, 1);

    // 7) q_ = relu(c @ Wq + bq)   [B,128]x[128,384]
    k_gemm_wmma<<<256, 256, 0, stream>>>(cbuf, Wq, bq, qbuf, B, 384, CDIM, 1);

    // 8) BAN fused logits
    k_ban_logits<<<B, 128, 0, stream>>>(vbar, qbuf, hmat, hbias, logit, HOUT);

    // 9) batch norm -> output
    k_batchnorm<<<CDIM, 256, 0, stream>>>(logit, gam, bet, (float*)d_out, B);
}